// mAPAt100_37074157699737
// MI455X (gfx1250) — compile-verified
//
#include <hip/hip_runtime.h>
#include <hip/hip_bf16.h>
#include <stdint.h>

#define MCOLS 2048
#define TPB   256
#define CPT   8   // columns per thread = MCOLS / TPB

typedef __attribute__((ext_vector_type(2))) float v2f;
typedef __attribute__((ext_vector_type(8))) float v8f;

__device__ __forceinline__ unsigned lds_offset(const void* p) {
    // generic pointer to __shared__: low 32 bits == wave-relative LDS byte address
    return (unsigned)(uintptr_t)p;
}

// Copy 32 bytes per lane (8 floats) global -> LDS via CDNA5 async path (ASYNCcnt).
__device__ __forceinline__ void async_row_copy(const float* gsrc, unsigned ldst) {
    asm volatile(
        "global_load_async_to_lds_b128 %0, %1, off\n\t"
        "global_load_async_to_lds_b128 %0, %1, off offset:16"
        :
        : "v"(ldst), "v"(gsrc)
        : "memory");
}

__global__ __launch_bounds__(TPB) void map_pass1(
        const float* __restrict__ rel,
        const int*   __restrict__ ranks,
        float*       __restrict__ ws,
        int Q, int NB)
{
    __shared__ __align__(16) float sbuf[2][MCOLS];   // double-buffered row (16 KB)
    __shared__ float wsum[TPB / 32];

    const int tid    = threadIdx.x;
    const int bid    = blockIdx.x;
    const int base_n = tid * CPT;

    // Per-thread column metadata is row-invariant: keep in registers.
    int   rr[CPT];
    float w [CPT];
#pragma unroll
    for (int j = 0; j < CPT; ++j) {
        rr[j] = ranks[base_n + j];
        w [j] = 1.0f / (float)(base_n + j + 1);
    }

    float acc = 0.0f;

    if (bid < Q) {
        async_row_copy(rel + (size_t)bid * MCOLS + base_n,
                       lds_offset(&sbuf[0][base_n]));
    }

    int b = 0;
    for (int r = bid; r < Q; r += NB) {
        asm volatile("s_wait_asynccnt 0x0" ::: "memory"); // my copies of row r done
        __syncthreads();                                  // all waves: buffer b ready
        const int rn = r + NB;
        if (rn < Q) {   // prefetch next row into other buffer (overlaps compute)
            async_row_copy(rel + (size_t)rn * MCOLS + base_n,
                           lds_offset(&sbuf[b ^ 1][base_n]));
        }
        const float* row = &sbuf[b][0];
        const float4* vp = (const float4*)(row + base_n);
        float4 v0 = vp[0];
        float4 v1 = vp[1];
        acc += v0.x * row[rr[0]] * w[0];
        acc += v0.y * row[rr[1]] * w[1];
        acc += v0.z * row[rr[2]] * w[2];
        acc += v0.w * row[rr[3]] * w[3];
        acc += v1.x * row[rr[4]] * w[4];
        acc += v1.y * row[rr[5]] * w[5];
        acc += v1.z * row[rr[6]] * w[6];
        acc += v1.w * row[rr[7]] * w[7];
        __syncthreads();                                  // all waves done reading b
        b ^= 1;
    }

    // Wave32 cross-lane reduction via V_WMMA_F32_16X16X4_F32:
    // A(16x4): lane L -> A[L%16, (L/16)*2] = acc (other K entries 0); B(4x16) = ones.
    // D[i,j] = acc_i + acc_{i+16}; summing a lane's 8 D rows + shfl_xor(16) = wave total.
    v2f a;  a[0]  = acc;  a[1]  = 0.0f;
    v2f bo; bo[0] = 1.0f; bo[1] = 1.0f;
    v8f c = {};
    c = __builtin_amdgcn_wmma_f32_16x16x4_f32(false, a, false, bo,
                                              (short)0, c, false, false);
    float s = ((c[0] + c[1]) + (c[2] + c[3])) + ((c[4] + c[5]) + (c[6] + c[7]));
    s += __shfl_xor(s, 16, 32);

    if ((tid & 31) == 0) wsum[tid >> 5] = s;
    __syncthreads();
    if (tid == 0) {
        float t = 0.0f;
#pragma unroll
        for (int i = 0; i < TPB / 32; ++i) t += wsum[i];
        ws[bid] = t;   // deterministic per-block partial
    }
}

__global__ __launch_bounds__(TPB) void map_finalize(
        const float* __restrict__ ws, float* __restrict__ out, int NB, int Q)
{
    __shared__ float red[TPB];
    const int tid = threadIdx.x;
    float s = 0.0f;
    for (int i = tid; i < NB; i += TPB) s += ws[i];   // fixed order
    red[tid] = s;
    __syncthreads();
    for (int off = TPB / 2; off > 0; off >>= 1) {
        if (tid < off) red[tid] += red[tid + off];
        __syncthreads();
    }
    if (tid == 0) {
        out[0] = red[0] / ((float)Q * (float)MCOLS);  // mAP
    }
}

__global__ __launch_bounds__(TPB) void map_fill(
        const int* __restrict__ ranks, float* __restrict__ out, int Q)
{
    __shared__ float scp;
    if (threadIdx.x == 0) {
        float cp = 0.0f;
        for (int i = 0; i < 100; ++i) cp += (float)ranks[i];
        scp = cp * 0.01f;
    }
    __syncthreads();
    const int idx = blockIdx.x * TPB + threadIdx.x;
    if (idx < Q) out[1 + idx] = scp;                  // cumulative_precision
}

extern "C" void kernel_launch(void* const* d_in, const int* in_sizes, int n_in,
                              void* d_out, int out_size, void* d_ws, size_t ws_size,
                              hipStream_t stream) {
    const float* rel   = (const float*)d_in[0];   // (Q, 2048) float32
    const int*   ranks = (const int*)d_in[1];     // (2048,)   int32
    float*       out   = (float*)d_out;           // [mAP, cp x Q]
    float*       ws    = (float*)d_ws;

    const int Q = in_sizes[0] / MCOLS;

    int NB = 2048;                                // block-partial count
    if ((size_t)NB * sizeof(float) > ws_size) NB = (int)(ws_size / sizeof(float));
    if (NB > Q) NB = Q;
    if (NB < 1) NB = 1;

    map_pass1   <<<NB, TPB, 0, stream>>>(rel, ranks, ws, Q, NB);
    map_finalize<<<1,  TPB, 0, stream>>>(ws, out, NB, Q);
    map_fill    <<<(Q + TPB - 1) / TPB, TPB, 0, stream>>>(ranks, out, Q);
}